// ResNetBlock_8890582303003
// MI455X (gfx1250) — compile-verified
//
#include <hip/hip_runtime.h>
#include <hip/hip_bf16.h>
#include <math.h>

// N=262144, C=128 fp32 ResNet block:
//   h1 = x@W1^T+b1 ; stats ; u = relu(a1*h1+c1)
//   h2 = u@W2^T+b2 ; stats ; out = relu(a2*h2+c2) + x
// h1/h2 live in d_out (in-place per-strip update in stage 2), ws holds only
// per-block stat partials + the 2x(a,c) coefficient vectors (~1MB).

#define kN      262144
#define kC      128
#define kStrips (kN / 16)   // 16384 strips of 16 rows
#define NB      1024        // gemm grid blocks
#define LST     132         // LDS row stride (floats): bank-conflict-free A reads
#define BN_EPS  1e-5f

typedef float v2f __attribute__((ext_vector_type(2)));
typedef float v4f __attribute__((ext_vector_type(4)));
typedef float v8f __attribute__((ext_vector_type(8)));

// One fused GEMM stage: out[n,:] = (transform(in[n,:])) @ W^T + bias, plus
// per-column sum / sum-of-squares partials per block.
// NORM=1 applies u = relu(a*in + c) (a=stats[0:128], c=stats[128:256]) while
// staging the strip to LDS. in/out may alias (same strip is read, barriered,
// then overwritten by the owning block only).
template <int NORM>
__global__ __launch_bounds__(256) void gemm_stage(
    const float* in, const float* __restrict__ W, const float* __restrict__ bias,
    const float* __restrict__ stats, float* out, float* __restrict__ partials) {
  __shared__ float lds[16 * LST];  // 16x128 strip, padded rows
  __shared__ float redS[256];
  __shared__ float redQ[256];

  const int tid  = threadIdx.x;
  const int lane = tid & 31;
  const int wave = tid >> 5;   // 8 waves: wave w -> output cols 16w..16w+15
  const int l16  = lane & 15;
  const int hi   = lane >> 4;  // half-wave select (K pairs 0,1 vs 2,3)
  const int col  = wave * 16 + l16;

  // B operand (Wᵀ) fragments held in registers for the whole kernel.
  // B[k][n] = W[n][k]; lane layout: lanes0-15 hold K=4kc+{0,1}, lanes16-31 K=4kc+{2,3}.
  v2f bfrag[32];
#pragma unroll
  for (int kc = 0; kc < 32; ++kc)
    bfrag[kc] = *(const v2f*)(W + col * kC + kc * 4 + 2 * hi);

  const float bn = bias[col];

  // Staging assignment: thread -> 8 contiguous floats of one row.
  const int srow = tid >> 4;
  const int scol = (tid & 15) * 8;

  v4f na0{}, na1{}, nc0{}, nc1{};
  if constexpr (NORM) {
    na0 = *(const v4f*)(stats + scol);
    na1 = *(const v4f*)(stats + scol + 4);
    nc0 = *(const v4f*)(stats + kC + scol);
    nc1 = *(const v4f*)(stats + kC + scol + 4);
  }

  // A-fragment LDS base: low lanes read k0,k0+1 ; high lanes k0+2,k0+3.
  // With LST=132, the 32-lane ds_load_b64 covers all 64 banks exactly once.
  const int aoff = l16 * LST + hi * 2;

  float csum = 0.f, csq = 0.f;

  for (int s = blockIdx.x; s < kStrips; s += NB) {
    const float* gin = in + (size_t)(s * 16 + srow) * kC + scol;
    v4f t0 = *(const v4f*)gin;
    v4f t1 = *(const v4f*)(gin + 4);
    if (s + NB < kStrips) __builtin_prefetch(gin + (size_t)NB * 16 * kC, 0, 0);
    if constexpr (NORM) {
#pragma unroll
      for (int j = 0; j < 4; ++j) {
        t0[j] = fmaxf(t0[j] * na0[j] + nc0[j], 0.f);
        t1[j] = fmaxf(t1[j] * na1[j] + nc1[j], 0.f);
      }
    }
    *(v4f*)(lds + srow * LST + scol)     = t0;
    *(v4f*)(lds + srow * LST + scol + 4) = t1;
    __syncthreads();

    v8f acc = {};
#pragma unroll
    for (int kc = 0; kc < 32; ++kc) {
      v2f a = *(const v2f*)(lds + aoff + kc * 4);
      acc = __builtin_amdgcn_wmma_f32_16x16x4_f32(false, a, false, bfrag[kc],
                                                  (short)0, acc, false, false);
    }

    // D layout: VGPR v -> row (hi*8+v), col = 16*wave + l16.
    float* gout = out + (size_t)(s * 16 + hi * 8) * kC + col;
#pragma unroll
    for (int v = 0; v < 8; ++v) {
      float h = acc[v] + bn;
      gout[(size_t)v * kC] = h;
      csum += h;
      csq += h * h;
    }
    __syncthreads();  // protect LDS before next staging (and in-place rewrite)
  }

  // Per-block column reduction: lane L and L+16 of wave w both hold column
  // 16w + (L&15); column index == tid for tid<128 after folding halves.
  redS[tid] = csum;
  redQ[tid] = csq;
  __syncthreads();
  if (tid < kC) {
    const int w = tid >> 4, l = tid & 15;
    partials[(size_t)tid * NB + blockIdx.x]        = redS[w * 32 + l] + redS[w * 32 + l + 16];
    partials[(size_t)(kC + tid) * NB + blockIdx.x] = redQ[w * 32 + l] + redQ[w * 32 + l + 16];
  }
}

// One block per column: fold NB partials -> a = g*rsqrt(var+eps), c = b - mean*a.
__global__ __launch_bounds__(256) void stats_kernel(
    const float* __restrict__ partials, const float* __restrict__ gamma,
    const float* __restrict__ beta, float* __restrict__ stats) {
  __shared__ float rs[256], rq[256];
  const int c = blockIdx.x, tid = threadIdx.x;
  float s = 0.f, q = 0.f;
  for (int b = tid; b < NB; b += 256) {
    s += partials[(size_t)c * NB + b];
    q += partials[(size_t)(kC + c) * NB + b];
  }
  rs[tid] = s;
  rq[tid] = q;
  __syncthreads();
  for (int off = 128; off > 0; off >>= 1) {
    if (tid < off) {
      rs[tid] += rs[tid + off];
      rq[tid] += rq[tid + off];
    }
    __syncthreads();
  }
  if (tid == 0) {
    const float inv_n = 1.0f / (float)kN;
    const float mean  = rs[0] * inv_n;
    const float var   = rq[0] * inv_n - mean * mean;
    const float a     = gamma[c] * rsqrtf(var + BN_EPS);
    stats[c]      = a;
    stats[kC + c] = beta[c] - mean * a;
  }
}

// out = relu(a2*h2 + c2) + x, in place over d_out.
__global__ __launch_bounds__(256) void finalize_kernel(
    float* hbuf, const float* __restrict__ x, const float* __restrict__ stats) {
  const size_t i = ((size_t)blockIdx.x * 256 + threadIdx.x) * 4;
  const int c = (int)(i & (kC - 1));
  v4f h  = *(v4f*)(hbuf + i);
  v4f xv = *(const v4f*)(x + i);
  v4f a  = *(const v4f*)(stats + c);
  v4f cc = *(const v4f*)(stats + kC + c);
  v4f r;
#pragma unroll
  for (int j = 0; j < 4; ++j) r[j] = fmaxf(h[j] * a[j] + cc[j], 0.f) + xv[j];
  *(v4f*)(hbuf + i) = r;
}

extern "C" void kernel_launch(void* const* d_in, const int* in_sizes, int n_in,
                              void* d_out, int out_size, void* d_ws, size_t ws_size,
                              hipStream_t stream) {
  const float* x      = (const float*)d_in[0];
  const float* W1     = (const float*)d_in[1];
  const float* b1     = (const float*)d_in[2];
  const float* gamma1 = (const float*)d_in[3];
  const float* beta1  = (const float*)d_in[4];
  const float* W2     = (const float*)d_in[5];
  const float* b2     = (const float*)d_in[6];
  const float* gamma2 = (const float*)d_in[7];
  const float* beta2  = (const float*)d_in[8];
  float* out = (float*)d_out;

  float* P      = (float*)d_ws;       // 2*kC*NB floats of stat partials
  float* stats1 = P + 2 * kC * NB;    // {a1[128], c1[128]}
  float* stats2 = stats1 + 2 * kC;    // {a2[128], c2[128]}

  // Stage 1: h1 = x@W1^T + b1 -> d_out, plus column stat partials.
  gemm_stage<0><<<NB, 256, 0, stream>>>(x, W1, b1, nullptr, out, P);
  stats_kernel<<<kC, 256, 0, stream>>>(P, gamma1, beta1, stats1);
  // Stage 2: u = relu(a1*h1+c1); h2 = u@W2^T + b2 -> d_out in place.
  gemm_stage<1><<<NB, 256, 0, stream>>>(out, W2, b2, stats1, out, P);
  stats_kernel<<<kC, 256, 0, stream>>>(P, gamma2, beta2, stats2);
  // out = relu(a2*h2+c2) + x
  finalize_kernel<<<(kN * kC) / 1024, 256, 0, stream>>>(out, x, stats2);
}